// GIN0_14516989460936
// MI455X (gfx1250) — compile-verified
//
#include <hip/hip_runtime.h>
#include <hip/hip_bf16.h>
#include <cstdint>

typedef float v2f __attribute__((ext_vector_type(2)));
typedef float v8f __attribute__((ext_vector_type(8)));

#define N_NODES   100000
#define N_EDGES   1600000
#define N_GRAPHS  128
#define HDIM      256
#define BN_EPS    1e-5f

// ---------------------------------------------------------------------------
// degree: deg[dst[e]] += 1
// ---------------------------------------------------------------------------
__global__ void deg_kernel(const int* __restrict__ dst, float* __restrict__ deg, int E) {
    int e = blockIdx.x * blockDim.x + threadIdx.x;
    if (e < E) atomicAdd(&deg[dst[e]], 1.0f);
}

// ---------------------------------------------------------------------------
// scatter-add: msg[dst[e], :] += X[src[e], :]   (d floats per edge, float4)
// shift = log2(d/4)
// ---------------------------------------------------------------------------
__global__ void edge_scatter(const float* __restrict__ X, const int* __restrict__ src,
                             const int* __restrict__ dst, float* __restrict__ msg,
                             unsigned total, int d, int shift) {
    unsigned gid = blockIdx.x * blockDim.x + threadIdx.x;
    if (gid >= total) return;
    unsigned e  = gid >> shift;
    unsigned c4 = (gid & ((1u << shift) - 1u)) << 2;
    int s = src[e], t = dst[e];
    float4 v = *(const float4*)(X + (size_t)s * d + c4);
    float* p = msg + (size_t)t * d + c4;
    atomicAdd(p + 0, v.x);
    atomicAdd(p + 1, v.y);
    atomicAdd(p + 2, v.z);
    atomicAdd(p + 3, v.w);
}

// ---------------------------------------------------------------------------
// combine: out = msg / max(deg,1) + X     (elementwise, float4)
// ---------------------------------------------------------------------------
__global__ void combine_kernel(const float* __restrict__ msg, const float* __restrict__ X,
                               const float* __restrict__ deg, float* __restrict__ out,
                               unsigned total, int shift) {
    unsigned i = blockIdx.x * blockDim.x + threadIdx.x;
    if (i >= total) return;
    unsigned r  = i >> shift;
    unsigned c4 = (i & ((1u << shift) - 1u)) << 2;
    int d = 4 << shift;
    float invd = 1.0f / fmaxf(deg[r], 1.0f);
    float4 m = *(const float4*)(msg + (size_t)r * d + c4);
    float4 x = *(const float4*)(X   + (size_t)r * d + c4);
    float4 o;
    o.x = m.x * invd + x.x;
    o.y = m.y * invd + x.y;
    o.z = m.z * invd + x.z;
    o.w = m.w * invd + x.w;
    *(float4*)(out + (size_t)r * d + c4) = o;
}

// ---------------------------------------------------------------------------
// C[M x 256] = relu(A[M x K] @ W[K x 256] + bias)
// Block: 256 threads (8 waves), tile 128 rows x 64 cols, K-chunk 32,
// double-buffered GLOBAL_LOAD_ASYNC_TO_LDS_B128 staging (ASYNCcnt pipelined).
// Wave w: rows 32*(w&3), cols 32*(w>>2); 2x2 grid of 16x16 f32 WMMA tiles.
// ---------------------------------------------------------------------------
__global__ __launch_bounds__(256)
void gemm_bias_relu(const float* __restrict__ A, const float* __restrict__ W,
                    const float* __restrict__ bias, float* __restrict__ C,
                    int M, int K) {
    // pad 36: row stride 144B keeps 16B alignment for async B128 LDS writes,
    // and stride-36 (mod 64 banks) keeps 16-row fragment reads conflict-free.
    __shared__ float As[2][128][36];
    __shared__ float Bs[2][32][64];

    const int tid  = threadIdx.x;
    const int lane = tid & 31;
    const int wave = tid >> 5;
    const int wrow = wave & 3;      // 0..3 -> row strip 32*wrow
    const int wcol = wave >> 2;     // 0..1 -> col strip 32*wcol
    const int half = lane >> 4;     // K-pair select (ISA f32 A/B layout)
    const int l16  = lane & 15;
    const int r0   = blockIdx.x * 128;
    const int c0   = blockIdx.y * 64;

    // Async loads are EXEC-masked: OOB rows never get written, so pre-zero the
    // A buffers once in the (single) partial block; zeros persist across chunks.
    if (r0 + 128 > M) {
        float* az = &As[0][0][0];
        for (int i = tid; i < 2 * 128 * 36; i += 256) az[i] = 0.0f;
    }
    __syncthreads();

    // stage(kc, p): issue async global->LDS B128 loads for the A and B tiles
    auto stage = [&](int kc, int p) {
        #pragma unroll
        for (int i = 0; i < 4; ++i) {           // A: 128x32 = 1024 float4
            int idx = tid + i * 256;
            int r   = idx >> 3;                 // 0..127
            int c4  = (idx & 7) << 2;           // 0..28
            int gr  = r0 + r;
            if (gr < M) {
                unsigned     lo = (unsigned)(uintptr_t)(void*)&As[p][r][c4];
                const float* ga = A + (size_t)gr * K + kc + c4;
                asm volatile("global_load_async_to_lds_b128 %0, %1, off"
                             :: "v"(lo), "v"(ga) : "memory");
            }
        }
        #pragma unroll
        for (int i = 0; i < 2; ++i) {           // B: 32x64 = 512 float4
            int idx = tid + i * 256;
            int r   = idx >> 4;                 // 0..31
            int c4  = (idx & 15) << 2;          // 0..60
            unsigned     lo = (unsigned)(uintptr_t)(void*)&Bs[p][r][c4];
            const float* ga = W + (size_t)(kc + r) * HDIM + c0 + c4;
            asm volatile("global_load_async_to_lds_b128 %0, %1, off"
                         :: "v"(lo), "v"(ga) : "memory");
        }
    };

    v8f acc[2][2];
    #pragma unroll
    for (int i = 0; i < 2; ++i)
        #pragma unroll
        for (int j = 0; j < 2; ++j)
            acc[i][j] = (v8f){0,0,0,0,0,0,0,0};

    // prologue: stage chunk 0 into buffer 0
    stage(0, 0);
    asm volatile("s_wait_asynccnt 0x0" ::: "memory");
    __syncthreads();

    for (int kc = 0; kc < K; kc += 32) {
        const int cur = (kc >> 5) & 1;
        if (kc + 32 < K) stage(kc + 32, cur ^ 1);   // prefetch next chunk (other buffer)

        #pragma unroll
        for (int k4 = 0; k4 < 8; ++k4) {
            const int kb = k4 * 4 + 2 * half;       // K-pair per half-wave
            v2f a0 = { As[cur][wrow * 32 +  0 + l16][kb], As[cur][wrow * 32 +  0 + l16][kb + 1] };
            v2f a1 = { As[cur][wrow * 32 + 16 + l16][kb], As[cur][wrow * 32 + 16 + l16][kb + 1] };
            v2f b0 = { Bs[cur][kb][wcol * 32 +  0 + l16], Bs[cur][kb + 1][wcol * 32 +  0 + l16] };
            v2f b1 = { Bs[cur][kb][wcol * 32 + 16 + l16], Bs[cur][kb + 1][wcol * 32 + 16 + l16] };
            acc[0][0] = __builtin_amdgcn_wmma_f32_16x16x4_f32(false, a0, false, b0, (short)0, acc[0][0], false, false);
            acc[0][1] = __builtin_amdgcn_wmma_f32_16x16x4_f32(false, a0, false, b1, (short)0, acc[0][1], false, false);
            acc[1][0] = __builtin_amdgcn_wmma_f32_16x16x4_f32(false, a1, false, b0, (short)0, acc[1][0], false, false);
            acc[1][1] = __builtin_amdgcn_wmma_f32_16x16x4_f32(false, a1, false, b1, (short)0, acc[1][1], false, false);
        }

        // prefetch must land before next iteration's compute; ds reads are
        // tracked by the compiler's own barrier waitcnts.
        asm volatile("s_wait_asynccnt 0x0" ::: "memory");
        __syncthreads();
    }

    // store: C/D layout -> VGPR v holds local row v + 8*half, col l16
    #pragma unroll
    for (int i = 0; i < 2; ++i) {
        #pragma unroll
        for (int j = 0; j < 2; ++j) {
            int gc = c0 + wcol * 32 + 16 * j + l16;
            float bv = bias[gc];
            #pragma unroll
            for (int v = 0; v < 8; ++v) {
                int gr = r0 + wrow * 32 + 16 * i + v + 8 * half;
                if (gr < M) {
                    float val = acc[i][j][v] + bv;
                    C[(size_t)gr * HDIM + gc] = fmaxf(val, 0.0f);
                }
            }
        }
    }
}

// ---------------------------------------------------------------------------
// BN stats: acc[c] += sum over row-chunk, acc[256+c] += sum of squares
// ---------------------------------------------------------------------------
__global__ __launch_bounds__(256)
void bn_stats(const float* __restrict__ H, float* __restrict__ acc, int M) {
    int c  = threadIdx.x;
    int r0 = blockIdx.x * 256;
    int r1 = min(r0 + 256, M);
    float s = 0.f, s2 = 0.f;
    for (int r = r0; r < r1; ++r) {
        float v = H[(size_t)r * HDIM + c];
        s += v; s2 += v * v;
    }
    atomicAdd(&acc[c], s);
    atomicAdd(&acc[HDIM + c], s2);
}

// ---------------------------------------------------------------------------
// BN apply (in place): h = g*(h-mu)*rsqrt(var+eps) + be ; biased variance
// ---------------------------------------------------------------------------
__global__ void bn_apply(float* __restrict__ H, const float* __restrict__ acc,
                         const float* __restrict__ g, const float* __restrict__ be,
                         unsigned total, float invM) {
    unsigned i = blockIdx.x * blockDim.x + threadIdx.x;
    if (i >= total) return;
    int c = i & (HDIM - 1);
    float mu  = acc[c] * invM;
    float var = acc[HDIM + c] * invM - mu * mu;
    float sc  = g[c] * rsqrtf(var + BN_EPS);
    H[i] = (H[i] - mu) * sc + be[c];
}

// ---------------------------------------------------------------------------
// global_add_pool: pooled[batch[r], :] += H[r, :]
// ---------------------------------------------------------------------------
__global__ void pool_kernel(const float* __restrict__ H, const int* __restrict__ batch,
                            float* __restrict__ pooled, unsigned total) {
    unsigned i = blockIdx.x * blockDim.x + threadIdx.x;
    if (i >= total) return;
    unsigned r  = i >> 6;           // 64 float4 per row (HDIM=256)
    unsigned c4 = (i & 63) << 2;
    int gidx = batch[r];
    float4 v = *(const float4*)(H + (size_t)r * HDIM + c4);
    float* p = pooled + (size_t)gidx * HDIM + c4;
    atomicAdd(p + 0, v.x);
    atomicAdd(p + 1, v.y);
    atomicAdd(p + 2, v.z);
    atomicAdd(p + 3, v.w);
}

// ---------------------------------------------------------------------------
// head: out[g] = relu(pooled[g] @ l1_w + l1_b) @ l2_w + l2_b
// one block per graph, one thread per hidden unit
// ---------------------------------------------------------------------------
__global__ __launch_bounds__(256)
void head_kernel(const float* __restrict__ pooled,
                 const float* __restrict__ w1, const float* __restrict__ b1,
                 const float* __restrict__ w2, const float* __restrict__ b2,
                 float* __restrict__ out) {
    __shared__ float red[256];
    int g = blockIdx.x, c = threadIdx.x;
    const float* pg = pooled + (size_t)g * HDIM;
    float s = b1[c];
    for (int k = 0; k < HDIM; ++k) s += pg[k] * w1[(size_t)k * HDIM + c];
    red[c] = fmaxf(s, 0.0f) * w2[c];
    __syncthreads();
    for (int off = 128; off > 0; off >>= 1) {
        if (c < off) red[c] += red[c + off];
        __syncthreads();
    }
    if (c == 0) out[g] = red[0] + b2[0];
}

// ---------------------------------------------------------------------------
extern "C" void kernel_launch(void* const* d_in, const int* in_sizes, int n_in,
                              void* d_out, int out_size, void* d_ws, size_t ws_size,
                              hipStream_t stream) {
    const float* x   = (const float*)d_in[0];
    const int*   ei  = (const int*)d_in[1];
    const int*   src = ei;
    const int*   dst = ei + N_EDGES;
    const int*   batch = (const int*)d_in[2];

    const float* W1[3] = { (const float*)d_in[3],  (const float*)d_in[9],  (const float*)d_in[15] };
    const float* B1[3] = { (const float*)d_in[4],  (const float*)d_in[10], (const float*)d_in[16] };
    const float* W2[3] = { (const float*)d_in[5],  (const float*)d_in[11], (const float*)d_in[17] };
    const float* B2[3] = { (const float*)d_in[6],  (const float*)d_in[12], (const float*)d_in[18] };
    const float* GG[3] = { (const float*)d_in[7],  (const float*)d_in[13], (const float*)d_in[19] };
    const float* BE[3] = { (const float*)d_in[8],  (const float*)d_in[14], (const float*)d_in[20] };
    const float* l1_w = (const float*)d_in[21];
    const float* l1_b = (const float*)d_in[22];
    const float* l2_w = (const float*)d_in[23];
    const float* l2_b = (const float*)d_in[24];
    float* out = (float*)d_out;

    // workspace carve-out
    char*  ws  = (char*)d_ws;
    size_t off = 0;
    auto take = [&](size_t bytes) -> void* {
        void* p = ws + off;
        off += (bytes + 255) & ~(size_t)255;
        return p;
    };
    float* deg    = (float*)take((size_t)N_NODES * 4);
    float* acc    = (float*)take((size_t)2 * HDIM * 4);
    float* pooled = (float*)take((size_t)N_GRAPHS * HDIM * 4);
    float* msg    = (float*)take((size_t)N_NODES * HDIM * 4);
    float* hA     = (float*)take((size_t)N_NODES * HDIM * 4);
    float* hT     = (float*)take((size_t)N_NODES * HDIM * 4);
    float* hH     = (float*)take((size_t)N_NODES * HDIM * 4);

    // degrees (same for all layers)
    hipMemsetAsync(deg, 0, (size_t)N_NODES * 4, stream);
    deg_kernel<<<(N_EDGES + 255) / 256, 256, 0, stream>>>(dst, deg, N_EDGES);

    const float invM = 1.0f / (float)N_NODES;
    dim3 gemm_grid((N_NODES + 127) / 128, HDIM / 64);

    for (int L = 0; L < 3; ++L) {
        const float* xin = (L == 0) ? x : hH;
        const int d      = (L == 0) ? 128 : HDIM;
        const int shift  = (L == 0) ? 5 : 6;       // log2(d/4)

        // mean aggregation + self
        hipMemsetAsync(msg, 0, (size_t)N_NODES * d * 4, stream);
        {
            unsigned total = (unsigned)N_EDGES << shift;
            edge_scatter<<<(total + 255) / 256, 256, 0, stream>>>(xin, src, dst, msg,
                                                                  total, d, shift);
        }
        {
            unsigned total = (unsigned)N_NODES << shift;
            combine_kernel<<<(total + 255) / 256, 256, 0, stream>>>(msg, xin, deg, hA,
                                                                    total, shift);
        }
        // MLP
        gemm_bias_relu<<<gemm_grid, 256, 0, stream>>>(hA, W1[L], B1[L], hT, N_NODES, d);
        gemm_bias_relu<<<gemm_grid, 256, 0, stream>>>(hT, W2[L], B2[L], hH, N_NODES, HDIM);
        // BatchNorm
        hipMemsetAsync(acc, 0, (size_t)2 * HDIM * 4, stream);
        bn_stats<<<(N_NODES + 255) / 256, 256, 0, stream>>>(hH, acc, N_NODES);
        {
            unsigned total = (unsigned)N_NODES * HDIM;
            bn_apply<<<(total + 255) / 256, 256, 0, stream>>>(hH, acc, GG[L], BE[L],
                                                              total, invM);
        }
    }

    // pooling + head
    hipMemsetAsync(pooled, 0, (size_t)N_GRAPHS * HDIM * 4, stream);
    {
        unsigned total = (unsigned)N_NODES * (HDIM / 4);
        pool_kernel<<<(total + 255) / 256, 256, 0, stream>>>(hH, batch, pooled, total);
    }
    head_kernel<<<N_GRAPHS, 256, 0, stream>>>(pooled, l1_w, l1_b, l2_w, l2_b, out);
}